// Model_14328010900113
// MI455X (gfx1250) — compile-verified
//
#include <hip/hip_runtime.h>
#include <math.h>

// ---------------------------------------------------------------------------
// Fused tiny-model kernel for MI455X (gfx1250, wave32).
// One block, one wave: serial conv-scan via lane shuffles, then three GEMMs
// on the CDNA5 matrix pipe using V_WMMA_F32_16X16X4_F32 (f32 == ref precision).
// ---------------------------------------------------------------------------

typedef __attribute__((ext_vector_type(2))) float v2f;
typedef __attribute__((ext_vector_type(8))) float v8f;

#define LOOP   100
#define BN_EPS 1e-5f

__device__ __forceinline__ float swishf(float v) {
    return v / (1.0f + __expf(-v));
}

// D = A(16x4,f32) * B(4x16,f32) + C(16x16,f32)
__device__ __forceinline__ v8f wmma4(v2f a, v2f b, v8f c) {
    // 8-arg pattern: (neg_a, A, neg_b, B, c_mod, C, reuse_a, reuse_b)
    return __builtin_amdgcn_wmma_f32_16x16x4_f32(
        false, a, false, b, (short)0, c, false, false);
}

__launch_bounds__(32)
__global__ void fused_model_kernel(
    const float* __restrict__ x,        // 16
    const float* __restrict__ conv_w,   // 9  (1,1,3,3)
    const float* __restrict__ conv_b,   // 1
    const float* __restrict__ bn_gamma, // 1
    const float* __restrict__ bn_beta,  // 1
    const float* __restrict__ bn_mean,  // 1
    const float* __restrict__ bn_var,   // 1
    const float* __restrict__ w1,       // 60x100
    const float* __restrict__ b1,       // 60
    const float* __restrict__ w2,       // 16x60
    const float* __restrict__ b2,       // 16
    const float* __restrict__ w3,       // 8x16
    const float* __restrict__ b3,       // 8
    float* __restrict__ out)            // 16x8
{
    // feats flat buffer: iteration t, pixel p stored at [t*16+p].
    // Read back as feats[i][k] = flat[i*100+k]  (the reshape is a pure
    // flat-index reinterpretation, so one array serves both phases).
    __shared__ float lds_feats[16 * 100];   // 1600 f
    __shared__ float lds_w1t[100 * 64];     // w1^T, N padded 60->64 (zeros)
    __shared__ float lds_w2t[64 * 16];      // w2^T, K padded 60->64 (zeros)
    __shared__ float lds_w3t[16 * 16];      // w3^T, N padded 8->16  (zeros)
    __shared__ float lds_h [16 * 64];       // GEMM1 output (cols 60..63 = 0)
    __shared__ float lds_h2[16 * 16];       // GEMM2 output

    const int lane  = threadIdx.x;          // 0..31 (one wave)
    const int ln16  = lane & 15;
    const int halfk = (lane >> 4) * 2;      // K sub-offset for A/B frags

    // ---- Stage weights into LDS (transposed + zero-padded) -----------------
    for (int idx = lane; idx < 100 * 64; idx += 32) {
        int k = idx >> 6, n = idx & 63;                 // [k*64 + n]
        lds_w1t[idx] = (n < 60) ? w1[n * 100 + k] : 0.0f;
    }
    for (int idx = lane; idx < 64 * 16; idx += 32) {
        int k = idx >> 4, n = idx & 15;                 // [k*16 + n]
        lds_w2t[idx] = (k < 60) ? w2[n * 60 + k] : 0.0f;
    }
    for (int idx = lane; idx < 16 * 16; idx += 32) {
        int k = idx >> 4, n = idx & 15;                 // [k*16 + n]
        lds_w3t[idx] = (n < 8) ? w3[n * 16 + k] : 0.0f;
    }

    // ---- Scalar params ------------------------------------------------------
    float cw[9];
#pragma unroll
    for (int q = 0; q < 9; ++q) cw[q] = conv_w[q];
    const float cbv     = conv_b[0];
    const float inv_std = bn_gamma[0] / sqrtf(bn_var[0] + BN_EPS);
    const float shift   = bn_beta[0] - bn_mean[0] * inv_std;

    // ---- Phase 1: serial conv scan, one pixel per lane (lanes 0..15) -------
    const int p = ln16;
    const int r = p >> 2, c = p & 3;
    int   nb_lane[9];
    float nb_mask[9];
#pragma unroll
    for (int i = 0; i < 3; ++i)
#pragma unroll
        for (int j = 0; j < 3; ++j) {
            int rr = r + i - 1, cc = c + j - 1;
            bool ok = (rr >= 0) & (rr < 4) & (cc >= 0) & (cc < 4);
            nb_lane[i * 3 + j] = ok ? (rr * 4 + cc) : 0;
            nb_mask[i * 3 + j] = ok ? 1.0f : 0.0f;
        }

    float xv = x[p];                        // lanes 16..31 mirror lanes 0..15
    for (int t = 0; t < LOOP; ++t) {
        float a = cbv;
#pragma unroll
        for (int q = 0; q < 9; ++q) {
            float nv = __shfl(xv, nb_lane[q], 32);
            a = fmaf(cw[q] * nb_mask[q], nv, a);
        }
        float b  = swishf(a) * inv_std + shift;
        float h  = a * b;
        float xn = copysignf(sqrtf(fabsf(h)), h);
        if (lane < 16) lds_feats[t * 16 + p] = xn;
        xv = xn;
    }
    __syncthreads();   // S_NOP for 1-wave WG, but a compiler LDS fence

    // ---- Phase 2: GEMM1  h1 = act(feats(16x100) @ w1^T + b1) ---------------
    // 25 K-steps x 4 N-tiles; A fragment reused across the 4 tiles per step.
    v8f acc0 = {}, acc1 = {}, acc2 = {}, acc3 = {};
    for (int k = 0; k < 100; k += 4) {
        v2f a;
        a.x = lds_feats[ln16 * 100 + k + halfk];
        a.y = lds_feats[ln16 * 100 + k + halfk + 1];
        const float* B0 = &lds_w1t[(k + halfk) * 64 + ln16];
        const float* B1 = &lds_w1t[(k + halfk + 1) * 64 + ln16];
        v2f bb0; bb0.x = B0[0];  bb0.y = B1[0];
        v2f bb1; bb1.x = B0[16]; bb1.y = B1[16];
        v2f bb2; bb2.x = B0[32]; bb2.y = B1[32];
        v2f bb3; bb3.x = B0[48]; bb3.y = B1[48];
        acc0 = wmma4(a, bb0, acc0);
        acc1 = wmma4(a, bb1, acc1);
        acc2 = wmma4(a, bb2, acc2);
        acc3 = wmma4(a, bb3, acc3);
    }
    // Epilogue: +b1, (swish-0.5)/0.5, zero pad cols >= 60, store to lds_h.
    {
        const int mbase = (lane >> 4) * 8;
#pragma unroll
        for (int v = 0; v < 8; ++v) {
            float vals[4] = { acc0[v], acc1[v], acc2[v], acc3[v] };
#pragma unroll
            for (int nt = 0; nt < 4; ++nt) {
                int col = nt * 16 + ln16;
                float bv  = b1[(col < 60) ? col : 59];
                float val = (col < 60)
                          ? (2.0f * swishf(vals[nt] + bv) - 1.0f)
                          : 0.0f;
                lds_h[(mbase + v) * 64 + col] = val;
            }
        }
    }
    __syncthreads();

    // ---- Phase 3: GEMM2  h2 = swish(h1(16x60) @ w2^T + b2), K padded to 64 --
    v8f accB = {};
    for (int k = 0; k < 64; k += 4) {
        v2f a;
        a.x = lds_h[ln16 * 64 + k + halfk];
        a.y = lds_h[ln16 * 64 + k + halfk + 1];
        v2f b;
        b.x = lds_w2t[(k + halfk) * 16 + ln16];
        b.y = lds_w2t[(k + halfk + 1) * 16 + ln16];
        accB = wmma4(a, b, accB);
    }
    {
        const int mbase = (lane >> 4) * 8;
        const float bv = b2[ln16];
#pragma unroll
        for (int v = 0; v < 8; ++v)
            lds_h2[(mbase + v) * 16 + ln16] = swishf(accB[v] + bv);
    }
    __syncthreads();

    // ---- Phase 4: GEMM3  y = h2(16x16) @ w3^T + b3, N padded 8->16 ----------
    v8f accC = {};
    for (int k = 0; k < 16; k += 4) {
        v2f a;
        a.x = lds_h2[ln16 * 16 + k + halfk];
        a.y = lds_h2[ln16 * 16 + k + halfk + 1];
        v2f b;
        b.x = lds_w3t[(k + halfk) * 16 + ln16];
        b.y = lds_w3t[(k + halfk + 1) * 16 + ln16];
        accC = wmma4(a, b, accC);
    }
    {
        const int mbase = (lane >> 4) * 8;
        if (ln16 < 8) {
            const float bv = b3[ln16];
#pragma unroll
            for (int v = 0; v < 8; ++v)
                out[(mbase + v) * 8 + ln16] = accC[v] + bv;
        }
    }
}

extern "C" void kernel_launch(void* const* d_in, const int* in_sizes, int n_in,
                              void* d_out, int out_size, void* d_ws, size_t ws_size,
                              hipStream_t stream) {
    (void)in_sizes; (void)n_in; (void)d_ws; (void)ws_size; (void)out_size;
    const float* x        = (const float*)d_in[0];
    const float* conv_w   = (const float*)d_in[1];
    const float* conv_b   = (const float*)d_in[2];
    const float* bn_gamma = (const float*)d_in[3];
    const float* bn_beta  = (const float*)d_in[4];
    const float* bn_mean  = (const float*)d_in[5];
    const float* bn_var   = (const float*)d_in[6];
    const float* w1       = (const float*)d_in[7];
    const float* b1       = (const float*)d_in[8];
    const float* w2       = (const float*)d_in[9];
    const float* b2       = (const float*)d_in[10];
    const float* w3       = (const float*)d_in[11];
    const float* b3       = (const float*)d_in[12];
    float* out            = (float*)d_out;

    fused_model_kernel<<<dim3(1), dim3(32), 0, stream>>>(
        x, conv_w, conv_b, bn_gamma, bn_beta, bn_mean, bn_var,
        w1, b1, w2, b2, w3, b3, out);
}